// LSTM_54047868453009
// MI455X (gfx1250) — compile-verified
//
#include <hip/hip_runtime.h>
#include <cstdint>
#include <cstddef>

// Problem constants (from reference): B=16384, T=1024, I=H=O=1
#define LSTM_B          16384
#define LSTM_T          1024
#define TCHUNK          64                    // time steps per TDM tile
#define WROWS           32                    // rows (batch elems) per wave == wave32
#define RSTRIDE         (TCHUNK + 1)          // 65 DWORDs/row in LDS (TDM pad => bank-conflict-free)
#define NCHUNK          (LSTM_T / TCHUNK)     // 16
#define WAVES_PER_BLOCK 4
#define BLOCK_THREADS   (WAVES_PER_BLOCK * 32)      // 128
#define SLAB_FLOATS     (2 * WROWS * RSTRIDE)       // per-wave double buffer (floats)
#define SHMEM_BYTES     (WAVES_PER_BLOCK * SLAB_FLOATS * sizeof(float))

typedef __attribute__((ext_vector_type(4))) unsigned int u32x4;
typedef __attribute__((ext_vector_type(8))) int          i32x8;
typedef __attribute__((ext_vector_type(4))) int          i32x4;

// gfx1250 has a hardware v_tanh_f32; fall back to exp2+rcp identity otherwise.
__device__ __forceinline__ float fast_tanh(float x) {
#if __has_builtin(__builtin_amdgcn_tanhf)
  return __builtin_amdgcn_tanhf(x);
#else
  // tanh(x) = 1 - 2/(exp2(2*log2e*x) + 1)
  float e = __builtin_amdgcn_exp2f(x * 2.8853900817779268f);
  return 1.0f - 2.0f * __builtin_amdgcn_rcpf(e + 1.0f);
#endif
}

#if __has_builtin(__builtin_amdgcn_tensor_load_to_lds)
#define LSTM_HAVE_TDM 1
// Issue one TDM 2D tile load: WROWS rows x TCHUNK cols of f32, row stride LSTM_T,
// into LDS at lds_byte with 1 DWORD of padding every 64 DWORDs (row stride 65 DW).
__device__ __forceinline__ void tdm_load_tile(uint32_t lds_byte, uint64_t gaddr) {
  // --- D# group 0 (128b): count=1 | lds_addr | global_addr[56:0] | type=2
  u32x4 g0 = { 1u,
               lds_byte,
               (uint32_t)gaddr,
               (uint32_t)((gaddr >> 32) & 0x1FFFFFFu) | (2u << 30) };
  // --- D# group 1 (256b):
  // w0: wg_mask=0 | data_size=2 (4B) | pad_enable=1 | pad_interval=5 (64 DW) | pad_amount=0 (1 DW)
  // tensor dims set huge (tile is always fully in-bounds; dims only gate OOB)
  i32x8 g1 = { (int)((2u << 16) | (1u << 20) | (5u << 22)),
               (int)0xFFFF0000u,                          // tensor_dim0[15:0]=0xFFFF
               (int)(0x7FFFu | (0xFFFFu << 16)),          // tensor_dim0[31:16] | tensor_dim1[15:0]
               (int)(0x7FFFu | ((unsigned)TCHUNK << 16)), // tensor_dim1[31:16] | tile_dim0=64
               (int)WROWS,                                // tile_dim1=32, tile_dim2=0 (unused)
               (int)LSTM_T,                               // tensor_dim0_stride = 1024 (lo32)
               0, 0 };                                    // stride hi / dim1_stride unused
  // --- D# groups 2/3: dims 2..4 unused (tile_dim3/4 = 0); keep tensor dims huge.
  i32x4 g2 = { 0x7FFFFFFF, 0x7FFFFFFF, 0, 0 };
  i32x4 g3 = { 0, (int)0xFFFF0000u, 0x7FFF, 0 };
  // clang-23 / therock toolchain: 6-arg form (u32x4, i32x8, i32x4, i32x4, i32x8, i32 cpol).
  // Extra 5th int32x8 group is zero-filled (unused for plain 2D, non-gather descriptors).
  i32x8 g4 = { 0, 0, 0, 0, 0, 0, 0, 0 };
  __builtin_amdgcn_tensor_load_to_lds(g0, g1, g2, g3, g4, /*cpol=*/0);
}
#else
#define LSTM_HAVE_TDM 0
#endif

__global__ void __launch_bounds__(BLOCK_THREADS)
lstm_scan_kernel(const float* __restrict__ x,
                 const float* __restrict__ W_ih,
                 const float* __restrict__ W_hh,
                 const float* __restrict__ b_ih,
                 const float* __restrict__ b_hh,
                 const float* __restrict__ W_out,
                 const float* __restrict__ b_out,
                 float* __restrict__ out) {
  extern __shared__ float smem[];
  const int tid  = threadIdx.x;
  const int wave = tid >> 5;
  const int lane = tid & 31;
  const int row  = blockIdx.x * BLOCK_THREADS + tid;  // this thread's batch element

  // Fold sigmoid(z)=0.5*tanh(z/2)+0.5 into pre-halved weights for gates i,f,o.
  const float wxi = 0.5f * W_ih[0], wxf = 0.5f * W_ih[1];
  const float wxg =        W_ih[2], wxo = 0.5f * W_ih[3];
  const float whi = 0.5f * W_hh[0], whf = 0.5f * W_hh[1];
  const float whg =        W_hh[2], who = 0.5f * W_hh[3];
  const float bi  = 0.5f * (b_ih[0] + b_hh[0]);
  const float bf  = 0.5f * (b_ih[1] + b_hh[1]);
  const float bg  =         b_ih[2] + b_hh[2];
  const float bo  = 0.5f * (b_ih[3] + b_hh[3]);
  const float wo  = W_out[0];
  const float bou = b_out[0];

  float h = 0.0f, c = 0.0f;

  auto step = [&](float xv) {
    float zi = fmaf(xv, wxi, fmaf(h, whi, bi));   // z/2 for sigmoid gates
    float zf = fmaf(xv, wxf, fmaf(h, whf, bf));
    float zg = fmaf(xv, wxg, fmaf(h, whg, bg));
    float zo = fmaf(xv, wxo, fmaf(h, who, bo));
    float ig = fmaf(0.5f, fast_tanh(zi), 0.5f);   // sigmoid(i)
    float fg = fmaf(0.5f, fast_tanh(zf), 0.5f);   // sigmoid(f)
    float gg = fast_tanh(zg);                     // tanh(g)
    float og = fmaf(0.5f, fast_tanh(zo), 0.5f);   // sigmoid(o)
    c = fmaf(fg, c, ig * gg);
    h = og * fast_tanh(c);
  };

#if LSTM_HAVE_TDM
  // Per-wave private slab: no cross-wave sync needed, TENSORcnt is per-wave.
  float* slab = smem + (size_t)wave * SLAB_FLOATS;
  const uint32_t lds0  = (uint32_t)(uintptr_t)slab;           // low 32b of flat addr == LDS offset
  const uint64_t gbase = (uint64_t)(uintptr_t)x +
                         (uint64_t)(blockIdx.x * BLOCK_THREADS + wave * 32) * (LSTM_T * 4ull);

  tdm_load_tile(lds0, gbase);                                  // chunk 0 -> buf 0
  for (int k = 0; k < NCHUNK - 1; ++k) {
    // prefetch chunk k+1 into the other buffer (async DMA, no VGPR traffic)
    tdm_load_tile(lds0 + (uint32_t)(((k + 1) & 1) * (WROWS * RSTRIDE * 4)),
                  gbase + (uint64_t)(k + 1) * (TCHUNK * 4ull));
    __builtin_amdgcn_s_wait_tensorcnt(1);                      // chunk k arrived (in-order)
    const float* r = slab + (size_t)((k & 1) * (WROWS * RSTRIDE)) + (size_t)lane * RSTRIDE;
#pragma unroll 4
    for (int j = 0; j < TCHUNK; ++j) step(r[j]);
  }
  __builtin_amdgcn_s_wait_tensorcnt(0);                        // last chunk arrived
  {
    const float* r = slab + (size_t)(((NCHUNK - 1) & 1) * (WROWS * RSTRIDE)) + (size_t)lane * RSTRIDE;
#pragma unroll 4
    for (int j = 0; j < TCHUNK; ++j) step(r[j]);
  }
#else
  // Fallback: direct vectorized row reads (strided across lanes; L2 absorbs it).
  const float4* xr = (const float4*)(x + (size_t)row * LSTM_T);
#pragma unroll 2
  for (int t4 = 0; t4 < LSTM_T / 4; ++t4) {
    float4 v = xr[t4];
    step(v.x); step(v.y); step(v.z); step(v.w);
  }
#endif

  // out = sigmoid(h*W_out + b_out) = 0.5*tanh(0.5*logit)+0.5
  float logit = fmaf(h, wo, bou);
  out[row] = fmaf(0.5f, fast_tanh(0.5f * logit), 0.5f);
}

extern "C" void kernel_launch(void* const* d_in, const int* in_sizes, int n_in,
                              void* d_out, int out_size, void* d_ws, size_t ws_size,
                              hipStream_t stream) {
  (void)in_sizes; (void)n_in; (void)out_size; (void)d_ws; (void)ws_size;
  const float* x     = (const float*)d_in[0];
  const float* W_ih  = (const float*)d_in[1];
  const float* W_hh  = (const float*)d_in[2];
  const float* b_ih  = (const float*)d_in[3];
  const float* b_hh  = (const float*)d_in[4];
  const float* W_out = (const float*)d_in[5];
  const float* b_out = (const float*)d_in[6];
  float* out = (float*)d_out;

  dim3 grid(LSTM_B / BLOCK_THREADS);   // 128 blocks
  dim3 block(BLOCK_THREADS);           // 128 threads = 4 waves
  lstm_scan_kernel<<<grid, block, SHMEM_BYTES, stream>>>(
      x, W_ih, W_hh, b_ih, b_hh, W_out, b_out, out);
}